// VMambaLayer_up_14224931685052
// MI455X (gfx1250) — compile-verified
//
#include <hip/hip_runtime.h>
#include <math.h>

// ---------------------------------------------------------------------------
// VMamba layer for gfx1250 (MI455X, wave32). fp32 end-to-end; GEMMs use
// V_WMMA_F32_16X16X4_F32 (fp32 WMMA). Scan is a 3-pass chunked recurrence.
// ---------------------------------------------------------------------------

typedef __attribute__((ext_vector_type(2))) float v2f;
typedef __attribute__((ext_vector_type(8))) float v8f;

#define LSEQ   9216          // 16*24*24 tokens per batch
#define NMTOT  (2 * LSEQ)    // total rows (B*L)
#define NCHK   96            // scan chunks
#define CLEN   96            // steps per chunk

__device__ __forceinline__ v8f wmma_f32_k4(v2f a, v2f b, v8f c) {
  // 8 args: (neg_a, A, neg_b, B, c_mod, C, reuse_a, reuse_b)
  return __builtin_amdgcn_wmma_f32_16x16x4_f32(false, a, false, b, (short)0, c,
                                               false, false);
}

__device__ __forceinline__ float siluf(float x) { return x / (1.f + expf(-x)); }
__device__ __forceinline__ float softplusf(float x) {
  return x > 0.f ? x + log1pf(expf(-x)) : log1pf(expf(x));
}

// ---------------------------------------------------------------------------
// Kernel 1: expand GEMM (16 rows x 96 @ 96x384, WMMA) + pixel shuffle + LN(48)
// grid 144 x 256. p = flat (b, h, w, d) over (2,8,12,12).
// ---------------------------------------------------------------------------
__global__ __launch_bounds__(256) void k_expand(
    const float* __restrict__ x, const float* __restrict__ expand_W,
    const float* __restrict__ pe_g, const float* __restrict__ pe_b,
    float* __restrict__ xseq) {
  __shared__ float aIn[16 * 96];
  __shared__ float buf[16 * 384];
  __shared__ float mu[128], rs[128];
  const int tid = threadIdx.x;
  const int p0 = blockIdx.x * 16;

  for (int j = tid; j < 16 * 96; j += 256) {
    int r = j / 96, k = j % 96;
    int p = p0 + r;
    int b = p / 1152, q = p % 1152;       // x is (B,96,8,12,12), q = h*144+w*12+d
    aIn[j] = x[(b * 96 + k) * 1152 + q];
  }
  __syncthreads();

  const int wave = tid >> 5, lane = tid & 31;
  const int half = lane >> 4, lm = lane & 15;
  for (int t = wave; t < 24; t += 8) {       // 24 N-tiles of 16
    const int n0 = t * 16;
    v8f acc = {};
    for (int kk = 0; kk < 96; kk += 4) {
      const int ka = kk + 2 * half;
      v2f av, bv;
      av.x = aIn[lm * 96 + ka];
      av.y = aIn[lm * 96 + ka + 1];
      bv.x = expand_W[ka * 384 + n0 + lm];
      bv.y = expand_W[(ka + 1) * 384 + n0 + lm];
      acc = wmma_f32_k4(av, bv, acc);
    }
    for (int r = 0; r < 8; ++r)
      buf[(r + 8 * half) * 384 + n0 + lm] = acc[r];
  }
  __syncthreads();

  if (tid < 128) {                            // LN stats: 16 rows x 8 groups of 48
    int row = tid >> 3, g = tid & 7;
    const float* v = &buf[row * 384 + g * 48];
    float s = 0.f, s2 = 0.f;
    for (int c = 0; c < 48; ++c) { s += v[c]; s2 += v[c] * v[c]; }
    float m = s * (1.f / 48.f);
    mu[tid] = m;
    rs[tid] = rsqrtf(s2 * (1.f / 48.f) - m * m + 1e-5f);
  }
  __syncthreads();

  for (int e = tid; e < 16 * 384; e += 256) { // normalize + pixel-shuffle scatter
    int row = e / 384, col = e % 384;
    int g = col / 48, c = col % 48;
    float val = (buf[e] - mu[row * 8 + g]) * rs[row * 8 + g] * pe_g[c] + pe_b[c];
    int p = p0 + row;
    int b = p / 1152, q = p % 1152;
    int h = q / 144, w = (q / 12) % 12, dd = q % 12;
    int i = col / 192, j = (col / 96) & 1, kb = (col / 48) & 1;
    int l = (2 * h + i) * 576 + (2 * w + j) * 24 + (2 * dd + kb);
    xseq[((long)b * LSEQ + l) * 96 + c] = val;
  }
}

// skip concat: xseq[b,l,48+c] = skip[b,c,l]   (skip is (2,48,16,24,24))
__global__ __launch_bounds__(256) void k_skip(const float* __restrict__ skip,
                                              float* __restrict__ xseq) {
  int idx = blockIdx.x * 256 + threadIdx.x;
  if (idx >= 2 * 48 * LSEQ) return;
  int b = idx / (48 * LSEQ);
  int rem = idx % (48 * LSEQ);
  int c = rem / LSEQ;
  int l = rem % LSEQ;
  xseq[((long)b * LSEQ + l) * 96 + 48 + c] = skip[idx];
}

// ---------------------------------------------------------------------------
// LN(96) + W_in GEMM: 16 rows x 96 @ 96x384 -> xi_raw(192) | z(192). WMMA.
// ---------------------------------------------------------------------------
__global__ __launch_bounds__(256) void k_ln_gemm_in(
    const float* __restrict__ xseq, const float* __restrict__ ln_g,
    const float* __restrict__ ln_b, const float* __restrict__ W_in,
    float* __restrict__ xiraw, float* __restrict__ zbuf) {
  __shared__ float hbuf[16 * 96];
  const int tid = threadIdx.x;
  const long m0 = (long)blockIdx.x * 16;

  for (int j = tid; j < 16 * 96; j += 256) hbuf[j] = xseq[m0 * 96 + j];
  __syncthreads();
  if (tid < 16) {
    float* v = &hbuf[tid * 96];
    float s = 0.f, s2 = 0.f;
    for (int c = 0; c < 96; ++c) { s += v[c]; s2 += v[c] * v[c]; }
    float m = s * (1.f / 96.f);
    float inv = rsqrtf(s2 * (1.f / 96.f) - m * m + 1e-5f);
    for (int c = 0; c < 96; ++c) v[c] = (v[c] - m) * inv * ln_g[c] + ln_b[c];
  }
  __syncthreads();

  const int wave = tid >> 5, lane = tid & 31;
  const int half = lane >> 4, lm = lane & 15;
  for (int t = wave; t < 24; t += 8) {
    const int n0 = t * 16;
    v8f acc = {};
    for (int kk = 0; kk < 96; kk += 4) {
      const int ka = kk + 2 * half;
      v2f av, bv;
      av.x = hbuf[lm * 96 + ka];
      av.y = hbuf[lm * 96 + ka + 1];
      bv.x = W_in[ka * 384 + n0 + lm];
      bv.y = W_in[(ka + 1) * 384 + n0 + lm];
      acc = wmma_f32_k4(av, bv, acc);
    }
    for (int r = 0; r < 8; ++r) {
      long m = m0 + r + 8 * half;
      int col = n0 + lm;
      if (col < 192) xiraw[m * 192 + col] = acc[r];
      else           zbuf[m * 192 + (col - 192)] = acc[r];
    }
  }
}

// ---------------------------------------------------------------------------
// Causal depthwise conv4 + SiLU, then dbc = xi@W_x (192->38),
// delta = softplus(dt@W_dt + b_dt). One block per token, 192 threads.
// ---------------------------------------------------------------------------
__global__ __launch_bounds__(192) void k_conv_dbc(
    const float* __restrict__ xiraw, const float* __restrict__ conv_w,
    const float* __restrict__ conv_b, const float* __restrict__ W_x,
    const float* __restrict__ W_dt, const float* __restrict__ b_dt,
    float* __restrict__ xiact, float* __restrict__ deltab,
    float* __restrict__ Bm, float* __restrict__ Cm) {
  __shared__ float xa[192];
  __shared__ float dbc[38];
  const int c = threadIdx.x;
  const long m = blockIdx.x;
  const int l = (int)(m % LSEQ);

  float s = conv_b[c];
  for (int k = 0; k < 4; ++k) {
    int ls = l - 3 + k;
    if (ls >= 0) s += conv_w[c * 4 + k] * xiraw[(m - 3 + k) * 192 + c];
  }
  float xv = siluf(s);
  xa[c] = xv;
  xiact[m * 192 + c] = xv;
  __syncthreads();
  if (c < 38) {
    float acc = 0.f;
    for (int k = 0; k < 192; ++k) acc += xa[k] * W_x[k * 38 + c];
    dbc[c] = acc;
  }
  __syncthreads();
  if (c < 16) {
    Bm[m * 16 + c] = dbc[6 + c];
    Cm[m * 16 + c] = dbc[22 + c];
  }
  float acc = b_dt[c];
  for (int r = 0; r < 6; ++r) acc += dbc[r] * W_dt[r * 192 + c];
  deltab[m * 192 + c] = softplusf(acc);
}

// ---------------------------------------------------------------------------
// Chunked scan. Lane = (2 channels x 16 states) per wave; block covers 16
// channels of one (batch, chunk). grid = 2*96*12.
// ---------------------------------------------------------------------------
__global__ __launch_bounds__(256) void k_scan1(
    const float* __restrict__ deltab, const float* __restrict__ xiact,
    const float* __restrict__ Bm, const float* __restrict__ A_log,
    float* __restrict__ carryA, float* __restrict__ carryH) {
  const int tid = threadIdx.x, lane = tid & 31;
  const int s = lane & 15, dloc = lane >> 4, wave = tid >> 5;
  const int bc = blockIdx.x;
  const int cb = (bc % 12) * 16;
  const int chunk = (bc / 12) % NCHK;
  const int b = bc / (12 * NCHK);
  const int d = cb + wave * 2 + dloc;
  const float A = -expf(A_log[d * 16 + s]);
  float aprod = 1.f, h = 0.f;
  const long base = (long)b * LSEQ + (long)chunk * CLEN;
  for (int t = 0; t < CLEN; ++t) {
    const long m = base + t;
    float dlt = deltab[m * 192 + d];
    float xv = xiact[m * 192 + d];
    float bv = Bm[m * 16 + s];
    float dA = expf(dlt * A);
    h = dA * h + dlt * xv * bv;
    aprod *= dA;
  }
  long slot = (long)(b * NCHK + chunk) * 3072 + d * 16 + s;
  carryA[slot] = aprod;
  carryH[slot] = h;
}

// chunk prefix: replaces carryH with the carry-IN state of each chunk
__global__ __launch_bounds__(256) void k_scan2(const float* __restrict__ carryA,
                                               float* __restrict__ carryH) {
  int idx = blockIdx.x * 256 + threadIdx.x;
  if (idx >= 2 * 3072) return;
  int b = idx / 3072, ds = idx % 3072;
  float run = 0.f;
  for (int ch = 0; ch < NCHK; ++ch) {
    long slot = (long)(b * NCHK + ch) * 3072 + ds;
    float a = carryA[slot];
    float hl = carryH[slot];
    carryH[slot] = run;          // carry-in for this chunk
    run = a * run + hl;
  }
}

// replay chunk with correct carry-in; reduce over states via shfl_xor (16-lane
// groups, wave32), add xi*Dp, gate with silu(z), write y_gated.
__global__ __launch_bounds__(256) void k_scan3(
    const float* __restrict__ deltab, const float* __restrict__ xiact,
    const float* __restrict__ Bm, const float* __restrict__ Cm,
    const float* __restrict__ A_log, const float* __restrict__ Dp,
    const float* __restrict__ zbuf, const float* __restrict__ carryH,
    float* __restrict__ ygated) {
  const int tid = threadIdx.x, lane = tid & 31;
  const int s = lane & 15, dloc = lane >> 4, wave = tid >> 5;
  const int bc = blockIdx.x;
  const int cb = (bc % 12) * 16;
  const int chunk = (bc / 12) % NCHK;
  const int b = bc / (12 * NCHK);
  const int d = cb + wave * 2 + dloc;
  const float A = -expf(A_log[d * 16 + s]);
  const float dpv = Dp[d];
  float h = carryH[(long)(b * NCHK + chunk) * 3072 + d * 16 + s];
  const long base = (long)b * LSEQ + (long)chunk * CLEN;
  for (int t = 0; t < CLEN; ++t) {
    const long m = base + t;
    float dlt = deltab[m * 192 + d];
    float xv = xiact[m * 192 + d];
    float bv = Bm[m * 16 + s];
    float dA = expf(dlt * A);
    h = dA * h + dlt * xv * bv;
    float contrib = h * Cm[m * 16 + s];
    contrib += __shfl_xor(contrib, 1);
    contrib += __shfl_xor(contrib, 2);
    contrib += __shfl_xor(contrib, 4);
    contrib += __shfl_xor(contrib, 8);
    if (s == 0) {
      float yv = contrib + xv * dpv;
      float zv = zbuf[m * 192 + d];
      ygated[m * 192 + d] = yv * siluf(zv);
    }
  }
}

// ---------------------------------------------------------------------------
// x_new = x + y_gated @ W_out (192 -> 96). WMMA, in-place on xseq.
// ---------------------------------------------------------------------------
__global__ __launch_bounds__(256) void k_gemm_out(
    const float* __restrict__ ygated, const float* __restrict__ W_out,
    float* __restrict__ xseq) {
  __shared__ float abuf[16 * 192];
  __shared__ float xres[16 * 96];
  const int tid = threadIdx.x;
  const long m0 = (long)blockIdx.x * 16;
  for (int j = tid; j < 16 * 192; j += 256) abuf[j] = ygated[m0 * 192 + j];
  for (int j = tid; j < 16 * 96; j += 256) xres[j] = xseq[m0 * 96 + j];
  __syncthreads();
  const int wave = tid >> 5, lane = tid & 31;
  const int half = lane >> 4, lm = lane & 15;
  if (wave < 6) {                      // 6 N-tiles of 16 (uniform per wave)
    const int n0 = wave * 16;
    v8f acc = {};
    for (int kk = 0; kk < 192; kk += 4) {
      const int ka = kk + 2 * half;
      v2f av, bv;
      av.x = abuf[lm * 192 + ka];
      av.y = abuf[lm * 192 + ka + 1];
      bv.x = W_out[ka * 96 + n0 + lm];
      bv.y = W_out[(ka + 1) * 96 + n0 + lm];
      acc = wmma_f32_k4(av, bv, acc);
    }
    for (int r = 0; r < 8; ++r) {
      int row = r + 8 * half;
      int col = n0 + lm;
      xseq[(m0 + row) * 96 + col] = acc[r] + xres[row * 96 + col];
    }
  }
}

// final projection 96 -> 48 (+bias). Output is raw-reshaped => flat (B,L,48).
__global__ __launch_bounds__(256) void k_final(
    const float* __restrict__ xseq, const float* __restrict__ W_cb,
    const float* __restrict__ b_cb, float* __restrict__ out) {
  int idx = blockIdx.x * 256 + threadIdx.x;
  if (idx >= NMTOT * 48) return;
  int m = idx / 48, c = idx % 48;
  float acc = b_cb[c];
  const float* row = &xseq[(long)m * 96];
  for (int k = 0; k < 96; ++k) acc += row[k] * W_cb[k * 48 + c];
  out[idx] = acc;
}

// ---------------------------------------------------------------------------
extern "C" void kernel_launch(void* const* d_in, const int* in_sizes, int n_in,
                              void* d_out, int out_size, void* d_ws,
                              size_t ws_size, hipStream_t stream) {
  (void)in_sizes; (void)n_in; (void)out_size; (void)ws_size;
  const float* x        = (const float*)d_in[0];
  const float* skip     = (const float*)d_in[1];
  const float* expand_W = (const float*)d_in[2];
  const float* pe_g     = (const float*)d_in[3];
  const float* pe_b     = (const float*)d_in[4];
  const float* ln_g     = (const float*)d_in[5];
  const float* ln_b     = (const float*)d_in[6];
  const float* W_in     = (const float*)d_in[7];
  const float* conv_w   = (const float*)d_in[8];
  const float* conv_b   = (const float*)d_in[9];
  const float* W_x      = (const float*)d_in[10];
  const float* W_dt     = (const float*)d_in[11];
  const float* b_dt     = (const float*)d_in[12];
  const float* A_log    = (const float*)d_in[13];
  const float* Dp       = (const float*)d_in[14];
  const float* W_out    = (const float*)d_in[15];
  const float* W_cb     = (const float*)d_in[16];
  const float* b_cb     = (const float*)d_in[17];

  float* ws = (float*)d_ws;
  float* xseq  = ws; ws += (long)NMTOT * 96;
  float* zbuf  = ws; ws += (long)NMTOT * 192;
  float* xiact = ws; ws += (long)NMTOT * 192;
  float* delt  = ws; ws += (long)NMTOT * 192;
  float* xiraw = ws; ws += (long)NMTOT * 192;   // reused as y_gated after conv
  float* Bm    = ws; ws += (long)NMTOT * 16;
  float* Cm    = ws; ws += (long)NMTOT * 16;
  float* carA  = ws; ws += 2L * NCHK * 3072;
  float* carH  = ws; ws += 2L * NCHK * 3072;

  k_expand<<<144, 256, 0, stream>>>(x, expand_W, pe_g, pe_b, xseq);
  k_skip<<<(2 * 48 * LSEQ + 255) / 256, 256, 0, stream>>>(skip, xseq);

  for (int i = 0; i < 2; ++i) {
    k_ln_gemm_in<<<NMTOT / 16, 256, 0, stream>>>(
        xseq, ln_g + i * 96, ln_b + i * 96, W_in + (long)i * 96 * 384,
        xiraw, zbuf);
    k_conv_dbc<<<NMTOT, 192, 0, stream>>>(
        xiraw, conv_w + (long)i * 192 * 4, conv_b + i * 192,
        W_x + (long)i * 192 * 38, W_dt + (long)i * 6 * 192, b_dt + i * 192,
        xiact, delt, Bm, Cm);
    k_scan1<<<2 * NCHK * 12, 256, 0, stream>>>(
        delt, xiact, Bm, A_log + (long)i * 192 * 16, carA, carH);
    k_scan2<<<24, 256, 0, stream>>>(carA, carH);
    k_scan3<<<2 * NCHK * 12, 256, 0, stream>>>(
        delt, xiact, Bm, Cm, A_log + (long)i * 192 * 16, Dp + i * 192, zbuf,
        carH, xiraw /* becomes y_gated */);
    k_gemm_out<<<NMTOT / 16, 256, 0, stream>>>(
        xiraw, W_out + (long)i * 192 * 96, xseq);
  }
  k_final<<<(NMTOT * 48 + 255) / 256, 256, 0, stream>>>(xseq, W_cb, b_cb,
                                                        (float*)d_out);
}